// GraphConv_84696755077586
// MI455X (gfx1250) — compile-verified
//
#include <hip/hip_runtime.h>
#include <math.h>

#define N_ENTITIES 100000
#define N_USERS    50000
#define N_ITEMS    30000
#define DIM        64
#define N_REL      16
#define N_CLUSTERS 4
#define N_EDGES    1500000
#define NNZ_IM     1000000
#define NNZ_ICM    1000000
#define N_HOPS     2

typedef __attribute__((ext_vector_type(2))) float v2f;
typedef __attribute__((ext_vector_type(8))) float v8f;

// ---------------------------------------------------------------------------
// rel_sum[d] = sum_r relation_emb[r][d]
// ---------------------------------------------------------------------------
__global__ void rel_sum_kernel(const float* __restrict__ rel, float* __restrict__ out) {
    int d = threadIdx.x;
    if (d < DIM) {
        float s = 0.f;
        #pragma unroll
        for (int r = 0; r < N_REL; ++r) s += rel[r * DIM + d];
        out[d] = s;
    }
}

// ---------------------------------------------------------------------------
// Row softmax of X[nRows,64] @ W[R,64]^T via V_WMMA_F32_16X16X4_F32.
// One wave computes a 16-row x 16-col logit tile (cols >= R zero-padded),
// then does a 16-lane shfl_xor softmax per row. Exact f32 math.
// A layout (16x4 f32): lane M = lane%16; VGPR0/1 = K pair (lane/16)*2 + {0,1}
// B layout mirrored with N in lanes. D: vgpr r -> M = r + 8*(lane/16), N = lane%16.
// ---------------------------------------------------------------------------
template <int R>
__global__ void row_softmax_wmma(const float* __restrict__ X,
                                 const float* __restrict__ W,
                                 float* __restrict__ out, int nRows) {
    int wave = (int)((blockIdx.x * (unsigned)blockDim.x + threadIdx.x) >> 5);
    int lane = threadIdx.x & 31;
    int rowBase = wave * 16;
    if (rowBase >= nRows) return;              // wave-uniform exit (EXEC stays all-1)

    const int n  = lane & 15;
    const int kh = (lane >> 4) * 2;            // 0 or 2
    const float* xrow = X + (size_t)(rowBase + n) * DIM;  // A: M = lane%16
    const float* wrow = (n < R) ? (W + (size_t)n * DIM) : nullptr;

    v8f acc = {};
    #pragma unroll
    for (int kk = 0; kk < 16; ++kk) {
        const int k0 = kk * 4 + kh;
        v2f a; a.x = xrow[k0]; a.y = xrow[k0 + 1];
        v2f b;
        if (n < R) { b.x = wrow[k0]; b.y = wrow[k0 + 1]; }
        else       { b.x = 0.f;      b.y = 0.f; }
        acc = __builtin_amdgcn_wmma_f32_16x16x4_f32(false, a, false, b,
                                                    (short)0, acc, false, false);
    }

    // Softmax per row over the first R columns.
    #pragma unroll
    for (int r = 0; r < 8; ++r) {
        const int m = r + ((lane >> 4) << 3);
        float v  = acc[r];
        float vm = (n < R) ? v : -1e30f;
        #pragma unroll
        for (int off = 1; off < 16; off <<= 1) vm = fmaxf(vm, __shfl_xor(vm, off, 32));
        float ev = (n < R) ? expf(v - vm) : 0.f;
        float s = ev;
        #pragma unroll
        for (int off = 1; off < 16; off <<= 1) s += __shfl_xor(s, off, 32);
        if (n < R) out[(size_t)(rowBase + m) * R + n] = ev / s;
    }
}

// ---------------------------------------------------------------------------
// Edge message + scatter: entity_agg[head] += e[tail] * relation_emb[etype]
//                                             * att[head,etype] * edge_imp
// One edge per wave; 32 lanes x float2 cover D=64.
// ---------------------------------------------------------------------------
__global__ void edge_msg_kernel(const float* __restrict__ e,
                                const float* __restrict__ rel,
                                const float* __restrict__ att,
                                const float* __restrict__ edge_imp,
                                const int* __restrict__ head,
                                const int* __restrict__ tail,
                                const int* __restrict__ etype,
                                float* __restrict__ agg, int nEdges) {
    long gid = (long)blockIdx.x * blockDim.x + threadIdx.x;
    int edge = (int)(gid >> 5);
    if (edge >= nEdges) return;
    int lane = (int)(gid & 31);
    int h = head[edge], t = tail[edge], r = etype[edge];
    float w = att[(size_t)h * N_REL + r] * edge_imp[edge];
    int d = lane * 2;
    v2f ev = *(const v2f*)(e   + (size_t)t * DIM + d);
    v2f rv = *(const v2f*)(rel + (size_t)r * DIM + d);
    float* dst = agg + (size_t)h * DIM + d;
    unsafeAtomicAdd(dst,     ev.x * rv.x * w);
    unsafeAtomicAdd(dst + 1, ev.y * rv.y * w);
}

// ---------------------------------------------------------------------------
// user_agg[rows[i]] += vals[i] * e[cols[i]]      (im pass)
// ---------------------------------------------------------------------------
__global__ void im_agg_kernel(const float* __restrict__ e,
                              const float* __restrict__ vals,
                              const int* __restrict__ rows,
                              const int* __restrict__ cols,
                              float* __restrict__ agg, int nnz) {
    long gid = (long)blockIdx.x * blockDim.x + threadIdx.x;
    int i = (int)(gid >> 5);
    if (i >= nnz) return;
    int lane = (int)(gid & 31);
    int rr = rows[i], c = cols[i];
    float w = vals[i];
    int d = lane * 2;
    v2f ev = *(const v2f*)(e + (size_t)c * DIM + d);
    float* dst = agg + (size_t)rr * DIM + d;
    unsafeAtomicAdd(dst,     ev.x * w);
    unsafeAtomicAdd(dst + 1, ev.y * w);
}

// ---------------------------------------------------------------------------
// Fused icm pass (segment_sum into dw + einsum('cud,uc->ud') collapsed):
// user_agg[rows[i]] += icm_vals[i] * ucls[rows[i], cls[i]] * e[cols[i]] * rel_sum
// ---------------------------------------------------------------------------
__global__ void icm_agg_kernel(const float* __restrict__ e,
                               const float* __restrict__ rel_sum,
                               const float* __restrict__ vals,
                               const float* __restrict__ ucls,
                               const int* __restrict__ cls,
                               const int* __restrict__ rows,
                               const int* __restrict__ cols,
                               float* __restrict__ agg, int nnz) {
    long gid = (long)blockIdx.x * blockDim.x + threadIdx.x;
    int i = (int)(gid >> 5);
    if (i >= nnz) return;
    int lane = (int)(gid & 31);
    int u = rows[i], c = cols[i], k = cls[i];
    float w = vals[i] * ucls[(size_t)u * N_CLUSTERS + k];
    int d = lane * 2;
    v2f ev = *(const v2f*)(e + (size_t)c * DIM + d);
    v2f rs = *(const v2f*)(rel_sum + d);
    float* dst = agg + (size_t)u * DIM + d;
    unsafeAtomicAdd(dst,     ev.x * rs.x * w);
    unsafeAtomicAdd(dst + 1, ev.y * rs.y * w);
}

// ---------------------------------------------------------------------------
// L2-normalize each row of agg -> cur, and res += cur. One row per wave.
// ---------------------------------------------------------------------------
__global__ void norm_acc_kernel(const float* __restrict__ agg,
                                float* __restrict__ cur,
                                float* __restrict__ res, int nRows) {
    int row = (int)((blockIdx.x * (unsigned)blockDim.x + threadIdx.x) >> 5);
    if (row >= nRows) return;
    int lane = threadIdx.x & 31;
    int d = lane * 2;
    v2f v = *(const v2f*)(agg + (size_t)row * DIM + d);
    float ss = v.x * v.x + v.y * v.y;
    #pragma unroll
    for (int off = 1; off < 32; off <<= 1) ss += __shfl_xor(ss, off, 32);
    float inv = 1.f / fmaxf(sqrtf(ss), 1e-12f);
    v2f o; o.x = v.x * inv; o.y = v.y * inv;
    *(v2f*)(cur + (size_t)row * DIM + d) = o;
    v2f rr = *(const v2f*)(res + (size_t)row * DIM + d);
    rr.x += o.x; rr.y += o.y;
    *(v2f*)(res + (size_t)row * DIM + d) = rr;
}

// ---------------------------------------------------------------------------
// cor_loss on disen_weight_att [4,16] — trivial, single thread.
// ---------------------------------------------------------------------------
__global__ void cor_loss_kernel(const float* __restrict__ dw, float* __restrict__ out) {
    float nt[N_CLUSTERS][N_REL];
    for (int i = 0; i < N_CLUSTERS; ++i) {
        float ss = 0.f;
        for (int k = 0; k < N_REL; ++k) { float v = dw[i * N_REL + k]; ss += v * v; }
        float inv = 1.f / fmaxf(sqrtf(ss), 1e-12f);
        for (int k = 0; k < N_REL; ++k) nt[i][k] = dw[i * N_REL + k] * inv;
    }
    float loss = 0.f;
    for (int i = 0; i < N_CLUSTERS; ++i) {
        float denom = 0.f, diag = 0.f;
        for (int j = 0; j < N_CLUSTERS; ++j) {
            float dd = 0.f;
            for (int k = 0; k < N_REL; ++k) dd += nt[i][k] * nt[j][k];
            float s = expf(dd * 5.0f);   // / TMP = *5
            denom += s;
            if (i == j) diag = s;
        }
        loss += logf(diag / denom);
    }
    out[0] = -loss;
}

// ---------------------------------------------------------------------------
extern "C" void kernel_launch(void* const* d_in, const int* in_sizes, int n_in,
                              void* d_out, int out_size, void* d_ws, size_t ws_size,
                              hipStream_t stream) {
    const float* user_emb     = (const float*)d_in[0];
    const float* entity_emb   = (const float*)d_in[1];
    const float* relation_emb = (const float*)d_in[3];
    const float* disen        = (const float*)d_in[4];
    const float* usr_cls_w    = (const float*)d_in[5];
    const float* edge_imp     = (const float*)d_in[6];
    const float* im_vals      = (const float*)d_in[7];
    const float* icm_vals     = (const float*)d_in[8];
    const int*   edge_index   = (const int*)d_in[9];   // [2, N_EDGES]
    const int*   edge_type    = (const int*)d_in[10];
    const int*   im_rows      = (const int*)d_in[11];
    const int*   im_cols      = (const int*)d_in[12];
    const int*   icm_cls      = (const int*)d_in[13];
    const int*   icm_rows     = (const int*)d_in[14];
    const int*   icm_cols     = (const int*)d_in[15];
    const int*   head = edge_index;
    const int*   tail = edge_index + N_EDGES;

    // Workspace layout (floats)
    float* ws      = (float*)d_ws;
    float* att     = ws;                                  // N_ENTITIES*N_REL
    float* e_cur   = att     + (size_t)N_ENTITIES * N_REL;
    float* u_cur   = e_cur   + (size_t)N_ENTITIES * DIM;
    float* ent_agg = u_cur   + (size_t)N_USERS    * DIM;
    float* usr_agg = ent_agg + (size_t)N_ENTITIES * DIM;
    float* ucls    = usr_agg + (size_t)N_USERS    * DIM;
    float* rel_sum = ucls    + (size_t)N_USERS    * N_CLUSTERS;

    float* ent_res = (float*)d_out;                          // [N_ENTITIES, DIM]
    float* usr_res = ent_res + (size_t)N_ENTITIES * DIM;     // [N_USERS, DIM]
    float* cor_out = usr_res + (size_t)N_USERS * DIM;        // [1]

    const size_t entBytes = (size_t)N_ENTITIES * DIM * sizeof(float);
    const size_t usrBytes = (size_t)N_USERS    * DIM * sizeof(float);

    hipMemcpyAsync(e_cur,   entity_emb, entBytes, hipMemcpyDeviceToDevice, stream);
    hipMemcpyAsync(u_cur,   user_emb,   usrBytes, hipMemcpyDeviceToDevice, stream);
    hipMemcpyAsync(ent_res, entity_emb, entBytes, hipMemcpyDeviceToDevice, stream);
    hipMemcpyAsync(usr_res, user_emb,   usrBytes, hipMemcpyDeviceToDevice, stream);

    rel_sum_kernel<<<1, 64, 0, stream>>>(relation_emb, rel_sum);

    const int entWaves = N_ENTITIES / 16;      // 6250 (exact)
    const int usrWaves = N_USERS / 16;         // 3125 (exact)

    for (int hop = 0; hop < N_HOPS; ++hop) {
        hipMemsetAsync(ent_agg, 0, entBytes, stream);
        hipMemsetAsync(usr_agg, 0, usrBytes, stream);

        row_softmax_wmma<N_REL><<<(entWaves + 7) / 8, 256, 0, stream>>>(
            e_cur, relation_emb, att, N_ENTITIES);
        row_softmax_wmma<N_CLUSTERS><<<(usrWaves + 7) / 8, 256, 0, stream>>>(
            u_cur, usr_cls_w, ucls, N_USERS);

        edge_msg_kernel<<<(int)(((long)N_EDGES * 32 + 255) / 256), 256, 0, stream>>>(
            e_cur, relation_emb, att, edge_imp, head, tail, edge_type, ent_agg, N_EDGES);

        im_agg_kernel<<<(int)(((long)NNZ_IM * 32 + 255) / 256), 256, 0, stream>>>(
            e_cur, im_vals, im_rows, im_cols, usr_agg, NNZ_IM);

        icm_agg_kernel<<<(int)(((long)NNZ_ICM * 32 + 255) / 256), 256, 0, stream>>>(
            e_cur, rel_sum, icm_vals, ucls, icm_cls, icm_rows, icm_cols, usr_agg, NNZ_ICM);

        norm_acc_kernel<<<(N_ENTITIES + 7) / 8, 256, 0, stream>>>(ent_agg, e_cur, ent_res, N_ENTITIES);
        norm_acc_kernel<<<(N_USERS + 7) / 8, 256, 0, stream>>>(usr_agg, u_cur, usr_res, N_USERS);
    }

    cor_loss_kernel<<<1, 1, 0, stream>>>(disen, cor_out);

    (void)in_sizes; (void)n_in; (void)out_size; (void)ws_size;
}